// EuclRiemGrassAtt_90220083020129
// MI455X (gfx1250) — compile-verified
//
#include <hip/hip_runtime.h>
#include <math.h>

// ---------------------------------------------------------------------------
// EuclRiemGrassAtt for MI455X (gfx1250), wave32 + V_WMMA_F32_16X16X4_F32.
// B=8, N=256, C=768, H=12, HD=64, 3H=36.
// d_out = out[B*N*C] ++ attn_[B*H*N*N]  (reference returns a tuple)
// ---------------------------------------------------------------------------

typedef float v2f __attribute__((ext_vector_type(2)));
typedef float v8f __attribute__((ext_vector_type(8)));

#define DEVFN __device__ __forceinline__

constexpr int Bsz = 8;
constexpr int Nn  = 256;
constexpr int Cc  = 768;
constexpr int Hh  = 12;
constexpr int HD  = 64;
constexpr int H3  = 36;       // 3*H
constexpr int C3  = 2304;     // 3*C
constexpr int BN  = 2048;     // B*N
constexpr int NN  = 65536;    // N*N (channel stride inside cat)
constexpr float LN_EPS = 1e-5f;

// ---- workspace layout (float offsets), total ~28.4M floats (~113 MB) ------
constexpr size_t SZ_QKV = (size_t)BN * C3;           // 4,718,592
constexpr size_t SZ_GR  = (size_t)Bsz * Hh * Nn * HD;// 1,572,864
constexpr size_t SZ_RS  = (size_t)Bsz * Hh * Nn;     // 24,576
constexpr size_t SZ_CAT = (size_t)Bsz * H3 * NN;     // 18,874,368
constexpr size_t OFF_QKV = 0;
constexpr size_t OFF_QGR = OFF_QKV + SZ_QKV;
constexpr size_t OFF_KGR = OFF_QGR + SZ_GR;
constexpr size_t OFF_SQ  = OFF_KGR + SZ_GR;
constexpr size_t OFF_SK  = OFF_SQ  + SZ_RS;
constexpr size_t OFF_FRO = OFF_SK  + SZ_RS;          // 96 floats
constexpr size_t OFF_LNS = OFF_FRO + 128;            // 16 floats (padded)
constexpr size_t OFF_CAT = OFF_LNS + 32;
constexpr size_t OFF_OHD = OFF_CAT + SZ_CAT;         // attn@v result

// ---- WMMA f32 16x16x4 wrapper ---------------------------------------------
// Fragment conventions (wave32):
//   A (16x4 f32): lane L holds M=L&15; K-pair = (L>=16 ? {2,3} : {0,1});
//                 .x = first K of pair, .y = second.
//   B (4x16 f32): lane L holds N-col = L&15; same K-pair striping as A.
//   C/D (16x16 f32): acc[r] in lane L is row r + (L>>4)*8, col L&15.
#if defined(__has_builtin)
#if __has_builtin(__builtin_amdgcn_wmma_f32_16x16x4_f32)
#define USE_WMMA_F32X4 1
#endif
#endif

DEVFN v8f wmma_f32_k4(v2f a, v2f b, v8f c) {
#ifdef USE_WMMA_F32X4
  // 8 args: (neg_a, A, neg_b, B, c_mod, C, reuse_a, reuse_b)
  return __builtin_amdgcn_wmma_f32_16x16x4_f32(false, a, false, b, (short)0, c,
                                               false, false);
#else
  // Placeholder so the file still compiles if the builtin is absent; the
  // instruction histogram (wmma==0) will flag this and we iterate.
  c[0] += a.x * b.x + a.y * b.y;
  return c;
#endif
}

// ---- wave32 / block reductions --------------------------------------------
DEVFN float wave_sum(float v) {
#pragma unroll
  for (int off = 16; off > 0; off >>= 1) v += __shfl_xor(v, off, 32);
  return v;
}
DEVFN float wave_max(float v) {
#pragma unroll
  for (int off = 16; off > 0; off >>= 1) v = fmaxf(v, __shfl_xor(v, off, 32));
  return v;
}
// blockDim.x == 256 (8 waves)
DEVFN float block_sum256(float v, float* red, int tid) {
  v = wave_sum(v);
  __syncthreads();
  if ((tid & 31) == 0) red[tid >> 5] = v;
  __syncthreads();
  float t = 0.f;
#pragma unroll
  for (int i = 0; i < 8; ++i) t += red[i];
  return t;
}
DEVFN float block_max256(float v, float* red, int tid) {
  v = wave_max(v);
  __syncthreads();
  if ((tid & 31) == 0) red[tid >> 5] = v;
  __syncthreads();
  float t = -3.4e38f;
#pragma unroll
  for (int i = 0; i < 8; ++i) t = fmaxf(t, red[i]);
  return t;
}

// ===========================================================================
// init: zero fro2[96] and lnstats[16]
// ===========================================================================
__global__ void k_init(float* __restrict__ fro2, float* __restrict__ lnstats) {
  int t = threadIdx.x;
  if (t < 96) fro2[t] = 0.f;
  if (t < 16) lnstats[t] = 0.f;
}

// ===========================================================================
// QKV GEMM: Y[M=2048, 2304] = X[2048,768] @ W[2304,768]^T + bias
// one wave per 16x64 tile; grid (128, 36)
// ===========================================================================
__global__ __launch_bounds__(32) void k_qkv_gemm(const float* __restrict__ X,
                                                 const float* __restrict__ W,
                                                 const float* __restrict__ bias,
                                                 float* __restrict__ Y) {
  const int lane = threadIdx.x;
  const int row0 = blockIdx.x * 16;
  const int col0 = blockIdx.y * 64;
  const int mi = lane & 15;
  const int hi = lane >> 4;
  const int kp = hi << 1;
  const float* xrow = X + (size_t)(row0 + mi) * Cc;
  v8f a0 = {}, a1 = {}, a2 = {}, a3 = {};
  for (int k0 = 0; k0 < Cc; k0 += 4) {
    if ((k0 & 63) == 0) __builtin_prefetch(xrow + k0 + 64, 0, 1);
    v2f a;
    a.x = xrow[k0 + kp];
    a.y = xrow[k0 + kp + 1];
    const float* wp = W + (size_t)(col0 + mi) * Cc + k0 + kp;
    v2f b0, b1, b2, b3;
    b0.x = wp[0];          b0.y = wp[1];
    b1.x = wp[16 * Cc];    b1.y = wp[16 * Cc + 1];
    b2.x = wp[32 * Cc];    b2.y = wp[32 * Cc + 1];
    b3.x = wp[48 * Cc];    b3.y = wp[48 * Cc + 1];
    a0 = wmma_f32_k4(a, b0, a0);
    a1 = wmma_f32_k4(a, b1, a1);
    a2 = wmma_f32_k4(a, b2, a2);
    a3 = wmma_f32_k4(a, b3, a3);
  }
#pragma unroll
  for (int r = 0; r < 8; ++r) {
    int row = row0 + r + hi * 8;
    float* yp = Y + (size_t)row * C3 + col0 + mi;
    yp[0]  = a0[r] + bias[col0 + mi];
    yp[16] = a1[r] + bias[col0 + 16 + mi];
    yp[32] = a2[r] + bias[col0 + 32 + mi];
    yp[48] = a3[r] + bias[col0 + 48 + mi];
  }
}

// ===========================================================================
// row sums of q and k over HD (for the covariance identity)
// sq/sk indexed [(b*H+h)*N + n]
// ===========================================================================
__global__ void k_rowsums(const float* __restrict__ qkv, float* __restrict__ sq,
                          float* __restrict__ sk) {
  int gid = blockIdx.x * blockDim.x + threadIdx.x;
  if (gid >= Bsz * Hh * Nn) return;
  int bh = gid >> 8, n = gid & 255;
  int b = bh / Hh, h = bh % Hh;
  const float* qp = qkv + (size_t)(b * Nn + n) * C3 + h * HD;
  const float* kp = qp + Cc;
  float s0 = 0.f, s1 = 0.f;
#pragma unroll
  for (int d = 0; d < HD; d += 4) {
    float4 q4 = *reinterpret_cast<const float4*>(qp + d);
    float4 k4 = *reinterpret_cast<const float4*>(kp + d);
    s0 += q4.x + q4.y + q4.z + q4.w;
    s1 += k4.x + k4.y + k4.z + k4.w;
  }
  sq[gid] = s0;
  sk[gid] = s1;
}

// ===========================================================================
// Reduced QR (modified Gram-Schmidt, in place) of a 256x64 matrix per (b,h).
// which: 0 -> q slice, 1 -> k slice of qkv. Matrix lives column-major in
// dynamic LDS (64*256 f32 = 64KB + 8-float reduction scratch; CDNA5 WGP has
// 320KB LDS). Every matrix element is owned by thread tid (= row index), so
// only the dot-product reductions need barriers.
// ===========================================================================
__global__ __launch_bounds__(256) void k_qr(const float* __restrict__ qkv,
                                            float* __restrict__ outQ,
                                            int which) {
  extern __shared__ float sm[];
  float* Mx  = sm;            // [64][256] column-major
  float* red = sm + HD * Nn;  // [8]
  const int tid = threadIdx.x;
  const int bh = blockIdx.x;
  const int b = bh / Hh, h = bh % Hh;
  const float* src = qkv + (size_t)b * Nn * C3 + which * Cc + h * HD;
  for (int i = tid; i < Nn * HD; i += 256) {
    int n = i >> 6, d = i & 63;
    Mx[d * Nn + n] = src[(size_t)n * C3 + d];
  }
  __syncthreads();
  for (int j = 0; j < HD; ++j) {
    float* cj = Mx + j * Nn;
    for (int i = 0; i < j; ++i) {
      const float* ci = Mx + i * Nn;
      float p = wave_sum(ci[tid] * cj[tid]);
      __syncthreads();
      if ((tid & 31) == 0) red[tid >> 5] = p;
      __syncthreads();
      float dot = red[0] + red[1] + red[2] + red[3] +
                  red[4] + red[5] + red[6] + red[7];
      cj[tid] -= dot * ci[tid];
    }
    float p = wave_sum(cj[tid] * cj[tid]);
    __syncthreads();
    if ((tid & 31) == 0) red[tid >> 5] = p;
    __syncthreads();
    float nrm = red[0] + red[1] + red[2] + red[3] +
                red[4] + red[5] + red[6] + red[7];
    cj[tid] *= rsqrtf(fmaxf(nrm, 1e-30f));
  }
  __syncthreads();
  float* dst = outQ + (size_t)bh * Nn * HD;  // [n][d] row-major
  for (int i = tid; i < Nn * HD; i += 256) {
    int n = i >> 6, d = i & 63;
    dst[i] = Mx[d * Nn + n];
  }
}

// ===========================================================================
// Scores: per (b,h) 16x16 tile, fused WMMA for S = q k^T and G = qgr kgr^T.
// cov tile derived from S: cov = (S - sq_n*sk_m/64)/63  (centering identity).
// Writes cat[b, h, :], cat[b, H+h, :] (raw cov), cat[b, 2H+h, :] and
// accumulates sum(cov^2) into fro2[b*H+h]. Grid (256 tiles, 96 bh), 32 thr.
// ===========================================================================
__global__ __launch_bounds__(32) void k_scores(
    const float* __restrict__ qkv, const float* __restrict__ qgr,
    const float* __restrict__ kgr, const float* __restrict__ sq,
    const float* __restrict__ sk, const float* __restrict__ p_scale,
    const float* __restrict__ p_gscale, float* __restrict__ cat,
    float* __restrict__ fro2) {
  const int lane = threadIdx.x;
  const int bh = blockIdx.y;
  const int b = bh / Hh, h = bh % Hh;
  const int n0 = (blockIdx.x >> 4) * 16;
  const int m0 = (blockIdx.x & 15) * 16;
  const int mi = lane & 15;
  const int hi = lane >> 4;
  const int kp = hi << 1;
  const float scale = p_scale[0];
  const float gscale = p_gscale[0];

  const float* Q = qkv + (size_t)b * Nn * C3 + h * HD;  // Q(n,d)=Q[n*C3+d]
  const float* K = Q + Cc;
  const float* qg = qgr + (size_t)bh * Nn * HD;
  const float* kg = kgr + (size_t)bh * Nn * HD;
  v8f accS = {}, accG = {};
  for (int k0 = 0; k0 < HD; k0 += 4) {
    v2f a, bb;
    a.x  = Q[(size_t)(n0 + mi) * C3 + k0 + kp];
    a.y  = Q[(size_t)(n0 + mi) * C3 + k0 + kp + 1];
    bb.x = K[(size_t)(m0 + mi) * C3 + k0 + kp];
    bb.y = K[(size_t)(m0 + mi) * C3 + k0 + kp + 1];
    accS = wmma_f32_k4(a, bb, accS);
    a.x  = qg[(n0 + mi) * HD + k0 + kp];
    a.y  = qg[(n0 + mi) * HD + k0 + kp + 1];
    bb.x = kg[(m0 + mi) * HD + k0 + kp];
    bb.y = kg[(m0 + mi) * HD + k0 + kp + 1];
    accG = wmma_f32_k4(a, bb, accG);
  }
  float* base = cat + (size_t)b * H3 * NN;
  float* c_attn = base + (size_t)h * NN;
  float* c_riem = base + (size_t)(Hh + h) * NN;
  float* c_gras = base + (size_t)(2 * Hh + h) * NN;
  const float inv63 = 1.0f / 63.0f, invHD = 1.0f / 64.0f;
  const float skm = sk[bh * Nn + m0 + mi];
  float fsum = 0.f;
#pragma unroll
  for (int r = 0; r < 8; ++r) {
    int n = n0 + r + hi * 8;
    int m = m0 + mi;
    float S = accS[r];
    float G = accG[r];
    float cov = (S - sq[bh * Nn + n] * skm * invHD) * inv63;
    c_attn[n * Nn + m] = S * scale;
    c_riem[n * Nn + m] = cov;
    c_gras[n * Nn + m] = G * G * gscale;
    fsum += cov * cov;
  }
  fsum = wave_sum(fsum);
  if (lane == 0) atomicAdd(&fro2[bh], fsum);
}

// ===========================================================================
// Frobenius-normalize the riem channels in place; accumulate per-batch
// LayerNorm sum/sumsq over the final cat values. Grid (1152, B), 256 thr.
// ===========================================================================
__global__ __launch_bounds__(256) void k_riem_lnstats(
    float* __restrict__ cat, const float* __restrict__ fro2,
    const float* __restrict__ p_rscale, float* __restrict__ lnstats) {
  const int b = blockIdx.y;
  const int tid = threadIdx.x;
  const float rscale = p_rscale[0];
  float* base = cat + (size_t)b * H3 * NN;
  float s = 0.f, s2 = 0.f;
  const int stride = gridDim.x * 256;
  for (int i = blockIdx.x * 256 + tid; i < H3 * NN; i += stride) {
    int ch = i >> 16;  // NN == 65536
    float x = base[i];
    if (ch >= Hh && ch < 2 * Hh) {
      x *= rscale * rsqrtf(fmaxf(fro2[b * Hh + (ch - Hh)], 1e-30f));
      base[i] = x;
    }
    s += x;
    s2 += x * x;
  }
  s = wave_sum(s);
  s2 = wave_sum(s2);
  if ((tid & 31) == 0) {
    atomicAdd(&lnstats[b * 2 + 0], s);
    atomicAdd(&lnstats[b * 2 + 1], s2);
  }
}

// ===========================================================================
// Fused LayerNorm + 1x1 conv (36->12) + row softmax. One block per (n, b);
// thread tid owns column m. 36x256 LDS tile. Writes attn_ to d_out tail.
// ===========================================================================
__global__ __launch_bounds__(256) void k_conv_softmax(
    const float* __restrict__ cat, const float* __restrict__ lnstats,
    const float* __restrict__ ln_g, const float* __restrict__ ln_b,
    const float* __restrict__ conv_w, const float* __restrict__ conv_b,
    float* __restrict__ attn_out) {
  __shared__ float smrow[H3][Nn];
  __shared__ float red[8];
  const int m = threadIdx.x;
  const int n = blockIdx.x;
  const int b = blockIdx.y;
  const float cnt = (float)H3 * (float)NN;
  const float mu = lnstats[b * 2 + 0] / cnt;
  const float var = lnstats[b * 2 + 1] / cnt - mu * mu;
  const float inv = rsqrtf(var + LN_EPS);
  const float* base = cat + (size_t)b * H3 * NN + (size_t)n * Nn + m;
  const float* gp = ln_g + (size_t)n * Nn + m;
  const float* bp = ln_b + (size_t)n * Nn + m;
#pragma unroll 4
  for (int ch = 0; ch < H3; ++ch) {
    float x = base[(size_t)ch * NN];
    smrow[ch][m] = (x - mu) * inv * gp[(size_t)ch * NN] + bp[(size_t)ch * NN];
  }
  __syncthreads();
  for (int o = 0; o < Hh; ++o) {
    float logit = conv_b[o];
#pragma unroll
    for (int c = 0; c < H3; ++c) logit += conv_w[o * H3 + c] * smrow[c][m];
    float mx = block_max256(logit, red, m);
    float e = __expf(logit - mx);
    float ssum = block_sum256(e, red, m);
    attn_out[((size_t)(b * Hh + o) * Nn + n) * Nn + m] = e / ssum;
  }
}

// ===========================================================================
// out_hd[b,h] = attn_[b,h] (256x256) @ v[b,h] (256x64). One wave per 16x64
// tile, grid (16, 96).
// ===========================================================================
__global__ __launch_bounds__(32) void k_av_gemm(const float* __restrict__ attn,
                                                const float* __restrict__ qkv,
                                                float* __restrict__ outhd) {
  const int lane = threadIdx.x;
  const int bh = blockIdx.y;
  const int b = bh / Hh, h = bh % Hh;
  const int n0 = blockIdx.x * 16;
  const int mi = lane & 15;
  const int hi = lane >> 4;
  const int kp = hi << 1;
  const float* P = attn + (size_t)bh * NN;                      // P(n,m)
  const float* V = qkv + (size_t)b * Nn * C3 + 2 * Cc + h * HD; // V(m,d)
  const float* prow = P + (size_t)(n0 + mi) * Nn;
  v8f a0 = {}, a1 = {}, a2 = {}, a3 = {};
  for (int k0 = 0; k0 < Nn; k0 += 4) {
    v2f a;
    a.x = prow[k0 + kp];
    a.y = prow[k0 + kp + 1];
    const float* vp = V + (size_t)(k0 + kp) * C3 + mi;
    v2f b0, b1, b2, b3;
    b0.x = vp[0];   b0.y = vp[C3];
    b1.x = vp[16];  b1.y = vp[C3 + 16];
    b2.x = vp[32];  b2.y = vp[C3 + 32];
    b3.x = vp[48];  b3.y = vp[C3 + 48];
    a0 = wmma_f32_k4(a, b0, a0);
    a1 = wmma_f32_k4(a, b1, a1);
    a2 = wmma_f32_k4(a, b2, a2);
    a3 = wmma_f32_k4(a, b3, a3);
  }
  float* dst = outhd + (size_t)bh * Nn * HD;
#pragma unroll
  for (int r = 0; r < 8; ++r) {
    int n = n0 + r + hi * 8;
    float* dp = dst + (size_t)n * HD + mi;
    dp[0] = a0[r];
    dp[16] = a1[r];
    dp[32] = a2[r];
    dp[48] = a3[r];
  }
}

// ===========================================================================
// Final projection: out[m=b*256+n, :768] = A(m,:) @ proj_w^T + proj_b where
// A(m, h*64+d) = outhd[b,h,n,d] (transpose folded into the fragment loads).
// Grid (128, 12).
// ===========================================================================
__global__ __launch_bounds__(32) void k_proj_gemm(
    const float* __restrict__ outhd, const float* __restrict__ W,
    const float* __restrict__ bias, float* __restrict__ Y) {
  const int lane = threadIdx.x;
  const int row0 = blockIdx.x * 16;
  const int col0 = blockIdx.y * 64;
  const int mi = lane & 15;
  const int hi = lane >> 4;
  const int kp = hi << 1;
  const int m = row0 + mi;
  const int b = m >> 8, n = m & 255;
  const float* abase = outhd + ((size_t)b * Hh * Nn + n) * HD;  // + h*N*HD + d
  v8f a0 = {}, a1 = {}, a2 = {}, a3 = {};
  for (int k0 = 0; k0 < Cc; k0 += 4) {
    const int kk = k0 + kp;
    const int hA = kk >> 6, dA = kk & 63;  // kk even, dA+1 stays in-head
    v2f a;
    a.x = abase[(size_t)hA * Nn * HD + dA];
    a.y = abase[(size_t)hA * Nn * HD + dA + 1];
    const float* wp = W + (size_t)(col0 + mi) * Cc + kk;
    v2f b0, b1, b2, b3;
    b0.x = wp[0];          b0.y = wp[1];
    b1.x = wp[16 * Cc];    b1.y = wp[16 * Cc + 1];
    b2.x = wp[32 * Cc];    b2.y = wp[32 * Cc + 1];
    b3.x = wp[48 * Cc];    b3.y = wp[48 * Cc + 1];
    a0 = wmma_f32_k4(a, b0, a0);
    a1 = wmma_f32_k4(a, b1, a1);
    a2 = wmma_f32_k4(a, b2, a2);
    a3 = wmma_f32_k4(a, b3, a3);
  }
#pragma unroll
  for (int r = 0; r < 8; ++r) {
    int row = row0 + r + hi * 8;
    float* yp = Y + (size_t)row * Cc + col0 + mi;
    yp[0]  = a0[r] + bias[col0 + mi];
    yp[16] = a1[r] + bias[col0 + 16 + mi];
    yp[32] = a2[r] + bias[col0 + 32 + mi];
    yp[48] = a3[r] + bias[col0 + 48 + mi];
  }
}

// ===========================================================================
extern "C" void kernel_launch(void* const* d_in, const int* in_sizes, int n_in,
                              void* d_out, int out_size, void* d_ws,
                              size_t ws_size, hipStream_t stream) {
  (void)in_sizes; (void)n_in; (void)out_size; (void)ws_size;
  const float* x      = (const float*)d_in[0];
  const float* qkv_w  = (const float*)d_in[1];
  const float* qkv_b  = (const float*)d_in[2];
  const float* proj_w = (const float*)d_in[3];
  const float* proj_b = (const float*)d_in[4];
  const float* scale  = (const float*)d_in[5];   // device scalar
  const float* rscale = (const float*)d_in[6];   // device scalar
  const float* gscale = (const float*)d_in[7];   // device scalar
  const float* ln_g   = (const float*)d_in[8];
  const float* ln_b   = (const float*)d_in[9];
  const float* conv_w = (const float*)d_in[10];
  const float* conv_b = (const float*)d_in[11];

  float* ws = (float*)d_ws;   // needs ~28.4M floats (~113.5 MB)
  float* qkv     = ws + OFF_QKV;
  float* qgr     = ws + OFF_QGR;
  float* kgr     = ws + OFF_KGR;
  float* sq      = ws + OFF_SQ;
  float* sk      = ws + OFF_SK;
  float* fro2    = ws + OFF_FRO;
  float* lnstats = ws + OFF_LNS;
  float* cat     = ws + OFF_CAT;
  float* outhd   = ws + OFF_OHD;

  float* out      = (float*)d_out;
  float* attn_out = out + (size_t)BN * Cc;   // tuple element 2

  k_init<<<1, 128, 0, stream>>>(fro2, lnstats);
  k_qkv_gemm<<<dim3(BN / 16, C3 / 64), 32, 0, stream>>>(x, qkv_w, qkv_b, qkv);
  k_rowsums<<<(Bsz * Hh * Nn) / 256, 256, 0, stream>>>(qkv, sq, sk);
  const size_t qr_lds = (size_t)(HD * Nn + 8) * sizeof(float);  // 64KB + red
  k_qr<<<Bsz * Hh, 256, qr_lds, stream>>>(qkv, qgr, 0);
  k_qr<<<Bsz * Hh, 256, qr_lds, stream>>>(qkv, kgr, 1);
  k_scores<<<dim3(256, Bsz * Hh), 32, 0, stream>>>(qkv, qgr, kgr, sq, sk, scale,
                                                   gscale, cat, fro2);
  k_riem_lnstats<<<dim3(1152, Bsz), 256, 0, stream>>>(cat, fro2, rscale,
                                                      lnstats);
  k_conv_softmax<<<dim3(Nn, Bsz), 256, 0, stream>>>(cat, lnstats, ln_g, ln_b,
                                                    conv_w, conv_b, attn_out);
  k_av_gemm<<<dim3(Nn / 16, Bsz * Hh), 32, 0, stream>>>(attn_out, qkv, outhd);
  k_proj_gemm<<<dim3(BN / 16, Cc / 64), 32, 0, stream>>>(outhd, proj_w, proj_b,
                                                         out);
}